// SpikingLorentzMultiHeadAttention_4492535791876
// MI455X (gfx1250) — compile-verified
//
#include <hip/hip_runtime.h>
#include <math.h>

typedef __attribute__((ext_vector_type(2))) float v2f;
typedef __attribute__((ext_vector_type(4))) float v4f;
typedef __attribute__((ext_vector_type(8))) float v8f;
typedef __attribute__((ext_vector_type(4))) int   v4i;

#define Bb 4
#define Tt 1024
#define Cc 1024
#define Hh 16
#define Dd 64

#if defined(__has_builtin)
#if __has_builtin(__builtin_amdgcn_global_load_async_to_lds_b128)
#define HAVE_ASYNC_LDS 1
#endif
#endif

#ifdef HAVE_ASYNC_LDS
// Builtin signature (from clang diagnostic): param 1 is a global-AS pointer
// to a 4 x i32 vector; LDS destination is an addrspace(3) pointer.
typedef __attribute__((address_space(1))) v4i* gv4i_p;
typedef __attribute__((address_space(3))) v4i* lv4i_p;
#define ASYNC_CP_B128(gptr, lptr)                                              \
    __builtin_amdgcn_global_load_async_to_lds_b128((gv4i_p)(gptr), (lv4i_p)(lptr), 0, 0)
#endif

// ---------------------------------------------------------------------------
// Kernel 1: spike gate.  spikes[b*T+t] = (sigmoid(x[b,t,:]·w_sur + b_sur) > thr)
// One wave (32 lanes) per row.
// ---------------------------------------------------------------------------
__global__ void spike_kernel(const float* __restrict__ x,
                             const float* __restrict__ w_sur,
                             const float* __restrict__ b_sur,
                             const float* __restrict__ thr,
                             float* __restrict__ spikes) {
    const int lane = threadIdx.x & 31;
    const int row  = blockIdx.x * (blockDim.x >> 5) + (threadIdx.x >> 5);
    if (row >= Bb * Tt) return;
    const float* xr = x + (size_t)row * Cc;
    float part = 0.f;
    for (int k = lane; k < Cc; k += 32) part += xr[k] * w_sur[k];
    part += __shfl_xor(part, 16, 32);
    part += __shfl_xor(part,  8, 32);
    part += __shfl_xor(part,  4, 32);
    part += __shfl_xor(part,  2, 32);
    part += __shfl_xor(part,  1, 32);
    if (lane == 0) {
        float z   = part + b_sur[0];
        float imp = 1.f / (1.f + expf(-z));
        spikes[row] = (imp > thr[0]) ? 1.f : 0.f;
    }
}

// ---------------------------------------------------------------------------
// Workgroup-tiled GEMM core: 4 waves compute a 64x64 tile of  A(MxK) @ W(KxN).
// B k-slices (16xN=64) are staged to LDS with async b128 copies (coalesced),
// each wave computes a 16x64 strip with 4 f32-WMMA accumulators, A fragments
// are loaded once and reused across the 4 N-subtiles.
//   NS     : row stride of W in floats
// Epilogues differ (scatter vs plain), so two kernels share this via macro.
// ---------------------------------------------------------------------------
#ifdef HAVE_ASYNC_LDS
#define STAGE_B128(g, l) ASYNC_CP_B128((g), (l))
#define STAGE_WAIT() asm volatile("s_wait_asynccnt 0x0" ::: "memory")
#else
#define STAGE_B128(g, l) (*(v4f*)(l) = *(const v4f*)(g))
#define STAGE_WAIT() ((void)0)
#endif

#define GEMM_TILE_BODY(Aptr, Wptr, NS)                                         \
    const int tid  = threadIdx.x;                                              \
    const int lane = tid & 31;                                                 \
    const int w    = tid >> 5;                                                 \
    const int lm   = lane & 15, lg = lane >> 4;                                \
    __shared__ float Bs[16][64];                                               \
    const int sr = tid >> 3;            /* staging row 0..15  */               \
    const int sc = (tid & 7) * 8;       /* staging col 0..56  */               \
    v8f acc[4] = {};                                                           \
    const float* aptr = (Aptr) + (size_t)(m0 + 16 * w + lm) * Cc + 2 * lg;     \
    for (int k0 = 0; k0 < Cc; k0 += 16) {                                      \
        const float* gsrc = (Wptr) + (size_t)(k0 + sr) * (NS) + n0 + sc;       \
        STAGE_B128(gsrc, &Bs[sr][sc]);                                         \
        STAGE_B128(gsrc + 4, &Bs[sr][sc + 4]);                                 \
        STAGE_WAIT();                                                          \
        __syncthreads();                                                       \
        _Pragma("unroll")                                                      \
        for (int ks = 0; ks < 4; ++ks) {                                       \
            v2f a = *(const v2f*)(aptr + k0 + ks * 4);                         \
            _Pragma("unroll")                                                  \
            for (int nt = 0; nt < 4; ++nt) {                                   \
                v2f b;                                                         \
                b.x = Bs[ks * 4 + 2 * lg    ][nt * 16 + lm];                   \
                b.y = Bs[ks * 4 + 2 * lg + 1][nt * 16 + lm];                   \
                acc[nt] = __builtin_amdgcn_wmma_f32_16x16x4_f32(               \
                    false, a, false, b, (short)0, acc[nt], false, false);      \
            }                                                                  \
        }                                                                      \
        __syncthreads();                                                       \
    }

// ---------------------------------------------------------------------------
// Kernel 2: QKV GEMM  (4096x1024)@(1024x3072)+bias, scattered into
// qkv[(which,b,h,t,d)] layout so q/k/v are (B,H,T,D)-contiguous.
// Grid: 64 M-tiles x 48 N-tiles = 3072 blocks of 128 threads.
// ---------------------------------------------------------------------------
__global__ __launch_bounds__(128) void qkv_gemm_kernel(
        const float* __restrict__ x, const float* __restrict__ W,
        const float* __restrict__ bias, float* __restrict__ qkv) {
    const int m0 = (blockIdx.x & 63) * 64;
    const int n0 = (blockIdx.x >> 6) * 64;
    GEMM_TILE_BODY(x, W, 3 * Cc)
#pragma unroll
    for (int nt = 0; nt < 4; ++nt) {
        const int col   = n0 + nt * 16 + lm;
        const float bs  = bias[col];
        const int which = col >> 10, rem = col & 1023;
        const int h = rem >> 6, d = rem & 63;
#pragma unroll
        for (int r = 0; r < 8; ++r) {
            const int row = m0 + 16 * w + r + 8 * lg;
            const int b_  = row >> 10, t = row & 1023;
            qkv[((size_t)((which * Bb + b_) * Hh + h) * Tt + t) * Dd + d] =
                acc[nt][r] + bs;
        }
    }
}

// ---------------------------------------------------------------------------
// Kernel 5: output projection (4096x1024)@(1024x1024)+bias, row-major out.
// Grid: 64 M-tiles x 16 N-tiles = 1024 blocks of 128 threads.
// ---------------------------------------------------------------------------
__global__ __launch_bounds__(128) void out_gemm_kernel(
        const float* __restrict__ y, const float* __restrict__ W,
        const float* __restrict__ bias, float* __restrict__ out) {
    const int m0 = (blockIdx.x & 63) * 64;
    const int n0 = (blockIdx.x >> 6) * 64;
    GEMM_TILE_BODY(y, W, Cc)
#pragma unroll
    for (int nt = 0; nt < 4; ++nt) {
        const int col  = n0 + nt * 16 + lm;
        const float bs = bias[col];
#pragma unroll
        for (int r = 0; r < 8; ++r)
            out[(size_t)(m0 + 16 * w + r + 8 * lg) * Cc + col] = acc[nt][r] + bs;
    }
}

// ---------------------------------------------------------------------------
// Kernel 3: in-place Lorentz expmap on q and k rows (64 floats each).
// Q' = [cosh(n), (sinh(n)/n)*u1..63],  K' = [cosh(n), -(sinh(n)/n)*u1..63]
// so that later  neg_inner = Q'·K'  is a plain 64-dot.
// One wave per row; rows 0..65535 are q, 65536..131071 are k.
// ---------------------------------------------------------------------------
__global__ void expmap_kernel(float* __restrict__ qkv) {
    const int lane = threadIdx.x & 31;
    const int wv   = blockIdx.x * (blockDim.x >> 5) + (threadIdx.x >> 5);
    if (wv >= 2 * Bb * Hh * Tt) return;
    const int which = wv >> 16;            // 0 = q, 1 = k
    float* base = qkv + (size_t)wv * Dd;
    v2f u = *(v2f*)(base + lane * 2);
    float part = u.x * u.x + u.y * u.y;
    if (lane == 0) part -= 2.f * u.x * u.x;   // mink = -u0^2 + sum(u1:^2)
    part += __shfl_xor(part, 16, 32);
    part += __shfl_xor(part,  8, 32);
    part += __shfl_xor(part,  4, 32);
    part += __shfl_xor(part,  2, 32);
    part += __shfl_xor(part,  1, 32);
    float nomin = sqrtf(fmaxf(part, 1e-8f));
    float tm  = coshf(nomin);
    float fac = sinhf(nomin) / nomin;
    float sgn = (which == 1) ? -1.f : 1.f;
    v2f o;
    o.x = (lane == 0) ? tm : sgn * fac * u.x;
    o.y = sgn * fac * u.y;
    *(v2f*)(base + lane * 2) = o;
}

// ---------------------------------------------------------------------------
// Kernel 4: flash-style causal Lorentz attention.
// One wave handles 16 query rows of one (b,h).  S-tile via 16 chained
// f32 WMMAs (K=64), arccosh^2 score transform, online softmax with shfl
// row-reductions, P transposed through LDS into A-fragments, P@V via WMMA.
// ---------------------------------------------------------------------------
__global__ __launch_bounds__(128) void attn_kernel(
        const float* __restrict__ qkv, const float* __restrict__ spikes,
        float* __restrict__ y) {
    const int lane = threadIdx.x & 31;
    const int wvin = threadIdx.x >> 5;
    const int wv   = blockIdx.x * 4 + wvin;
    const int ib = wv & 63;               // query tile index (T/16 = 64)
    const int bh = wv >> 6;
    const int b  = bh >> 4, h = bh & 15;
    const int lm = lane & 15, lg = lane >> 4;

    const float* qp = qkv + (size_t)((0 * Bb + b) * Hh + h) * Tt * Dd;
    const float* kp = qkv + (size_t)((1 * Bb + b) * Hh + h) * Tt * Dd;
    const float* vp = qkv + (size_t)((2 * Bb + b) * Hh + h) * Tt * Dd;

    __shared__ float ldsP[4][16 * 17];
    float* myP = ldsP[wvin];

    // Preload Q' A-fragments for all 16 K-steps (16x64 tile).
    v2f aq[16];
#pragma unroll
    for (int ks = 0; ks < 16; ++ks)
        aq[ks] = *(const v2f*)(qp + (size_t)(ib * 16 + lm) * Dd + ks * 4 + lg * 2);

    float m_acc[8], l_acc[8];
    v8f o[4] = {};
#pragma unroll
    for (int r = 0; r < 8; ++r) { m_acc[r] = -INFINITY; l_acc[r] = 0.f; }

    for (int jb = 0; jb <= ib; ++jb) {
        // S = Q' * K'^T  (16x16 tile, K=64)
        v8f s = {};
#pragma unroll
        for (int ks = 0; ks < 16; ++ks) {
            v2f bk = *(const v2f*)(kp + (size_t)(jb * 16 + lm) * Dd + ks * 4 + lg * 2);
            s = __builtin_amdgcn_wmma_f32_16x16x4_f32(false, aq[ks], false, bk,
                                                      (short)0, s, false, false);
        }
        const int jcol = jb * 16 + lm;
        float p[8];
#pragma unroll
        for (int r = 0; r < 8; ++r) {
            const int irow = ib * 16 + r + 8 * lg;
            float sc;
            if (jcol <= irow) {
                float ni   = fmaxf(s[r], 1.0f + 1e-7f);
                float dist = acoshf(ni);
                sc = -dist * dist * 0.125f;          // 1/sqrt(D)=1/8
            } else {
                sc = -INFINITY;
            }
            // row max over the 16-lane group holding this row
            float rm = sc;
            rm = fmaxf(rm, __shfl_xor(rm, 8, 32));
            rm = fmaxf(rm, __shfl_xor(rm, 4, 32));
            rm = fmaxf(rm, __shfl_xor(rm, 2, 32));
            rm = fmaxf(rm, __shfl_xor(rm, 1, 32));
            float mnew  = fmaxf(m_acc[r], rm);
            float scale = __expf(m_acc[r] - mnew);
            float pe    = __expf(sc - mnew);
            float rs = pe;
            rs += __shfl_xor(rs, 8, 32);
            rs += __shfl_xor(rs, 4, 32);
            rs += __shfl_xor(rs, 2, 32);
            rs += __shfl_xor(rs, 1, 32);
            l_acc[r] = l_acc[r] * scale + rs;
            m_acc[r] = mnew;
#pragma unroll
            for (int nt = 0; nt < 4; ++nt) o[nt][r] *= scale;
            p[r] = pe;
        }
        // Transpose P (C-layout -> A-layout) through per-wave LDS.
#pragma unroll
        for (int r = 0; r < 8; ++r) myP[(r + 8 * lg) * 17 + lm] = p[r];
        asm volatile("s_wait_dscnt 0x0" ::: "memory");
        v2f pa[4];
#pragma unroll
        for (int ks = 0; ks < 4; ++ks) {
            v2f t;
            t.x = myP[lm * 17 + ks * 4 + lg * 2];
            t.y = myP[lm * 17 + ks * 4 + lg * 2 + 1];
            pa[ks] = t;
        }
        asm volatile("s_wait_dscnt 0x0" ::: "memory");
        // O += P @ V_block   (16x16 @ 16x64)
#pragma unroll
        for (int ks = 0; ks < 4; ++ks) {
            const int krow = jb * 16 + ks * 4 + lg * 2;
#pragma unroll
            for (int nt = 0; nt < 4; ++nt) {
                v2f bv;
                bv.x = vp[(size_t)krow * Dd + nt * 16 + lm];
                bv.y = vp[(size_t)(krow + 1) * Dd + nt * 16 + lm];
                o[nt] = __builtin_amdgcn_wmma_f32_16x16x4_f32(false, pa[ks], false, bv,
                                                              (short)0, o[nt], false, false);
            }
        }
    }
    // Epilogue: normalize, apply spike mask, write y[b, t, h*64+d].
#pragma unroll
    for (int r = 0; r < 8; ++r) {
        const int irow = ib * 16 + r + 8 * lg;
        float sp  = spikes[b * Tt + irow];
        float inv = sp / l_acc[r];
#pragma unroll
        for (int nt = 0; nt < 4; ++nt)
            y[(size_t)(b * Tt + irow) * Cc + h * Dd + nt * 16 + lm] = o[nt][r] * inv;
    }
}

// ---------------------------------------------------------------------------
// Host-side launcher.
// Workspace layout (floats): qkv[3*B*H*T*D] | y[B*T*C] | spikes[B*T]
//   = 12,582,912 + 4,194,304 + 4,096 floats  (~67 MB)
// ---------------------------------------------------------------------------
extern "C" void kernel_launch(void* const* d_in, const int* in_sizes, int n_in,
                              void* d_out, int out_size, void* d_ws, size_t ws_size,
                              hipStream_t stream) {
    const float* x     = (const float*)d_in[0];
    const float* W_qkv = (const float*)d_in[1];
    const float* b_qkv = (const float*)d_in[2];
    const float* W_out = (const float*)d_in[3];
    const float* b_out = (const float*)d_in[4];
    const float* w_sur = (const float*)d_in[5];
    const float* b_sur = (const float*)d_in[6];
    const float* thr   = (const float*)d_in[7];
    float* out = (float*)d_out;

    float* qkv    = (float*)d_ws;                    // 3*B*H*T*D
    float* yb     = qkv + (size_t)3 * Bb * Hh * Tt * Dd;
    float* spikes = yb + (size_t)Bb * Tt * Cc;

    // 1) spike gate: 4096 rows, 8 waves/block
    spike_kernel<<<512, 256, 0, stream>>>(x, w_sur, b_sur, thr, spikes);
    // 2) QKV GEMM: 64 x 48 tiles of 64x64, 128 threads/block
    qkv_gemm_kernel<<<3072, 128, 0, stream>>>(x, W_qkv, b_qkv, qkv);
    // 3) expmap on q and k: 131072 rows, 8 waves/block
    expmap_kernel<<<16384, 256, 0, stream>>>(qkv);
    // 4) attention: 64 tiles * 64 (b,h) = 4096 waves, 4 waves/block
    attn_kernel<<<1024, 128, 0, stream>>>(qkv, spikes, yb);
    // 5) output projection: 64 x 16 tiles of 64x64
    out_gemm_kernel<<<1024, 128, 0, stream>>>(yb, W_out, b_out, out);
}